// VectorQuantization_85985245266491
// MI455X (gfx1250) — compile-verified
//
#include <hip/hip_runtime.h>

// VectorQuantization on gfx1250:
//   x: [b=4, n=2048, 8*64] fp32, means: [8, 1024, 64] fp32
//   out[b,n,h] = argmin_k ||x - means_k||^2
// Cross term via V_WMMA_F32_16X16X4_F32 (fp32 keeps the argmin exact).
// ||x||^2 is constant in k -> dropped from the argmin entirely.
// ||m_k||^2 precomputed once per block into LDS.

#define NHEADS  8
#define DIMH    64
#define NCLUST  1024
#define TOKTILE 16
#define CT      32                 // clusters per main-loop iteration (2 subtiles of 16)
#define NT      (NCLUST / CT)      // 32 iterations
#define WAVES   8
#define LSTRIDE 68                 // padded row stride (floats) for LDS bank spread

typedef __attribute__((ext_vector_type(2))) float v2f;
typedef __attribute__((ext_vector_type(8))) float v8f;

__global__ __launch_bounds__(256) void vq_argmin_kernel(
    const float* __restrict__ x,
    const float* __restrict__ means,
    float* __restrict__ out) {

  __shared__ float lds_b[2][CT * LSTRIDE];   // double-buffered means tile
  __shared__ float lds_msq[NCLUST];          // ||m_c||^2 for the whole head

  const int tid  = threadIdx.x;
  const int lane = tid & 31;
  const int wave = tid >> 5;
  const int head = blockIdx.y;
  const int token0 = (blockIdx.x * WAVES + wave) * TOKTILE;

  const int nlo = lane & 15;        // A row / B,C column within 16
  const int hi  = (lane >> 4) & 1;  // wave half

  const float* mhead = means + (size_t)head * NCLUST * DIMH;

  // ---------------- A fragments: 16 tokens x 64 dims, 16 chunks of K=4 ----
  // A 16x4 layout: lanes 0-15 -> M=lane, K={0,1}; lanes 16-31 -> M=lane-16, K={2,3}
  const float* xrow = x + (size_t)(token0 + nlo) * (NHEADS * DIMH) + (size_t)head * DIMH;
  v2f a_frag[16];
#pragma unroll
  for (int kk = 0; kk < 16; ++kk) {
    const float* p = xrow + kk * 4 + hi * 2;
    v2f a; a.x = p[0]; a.y = p[1];
    a_frag[kk] = a;
  }

  // ---------------- precompute ||m_c||^2 for all 1024 clusters ------------
  for (int c = tid; c < NCLUST; c += 256) {
    const float* mp = mhead + (size_t)c * DIMH;
    float s = 0.f;
#pragma unroll
    for (int q = 0; q < 16; ++q) {
      float4 v = *(const float4*)(mp + q * 4);
      s += v.x * v.x + v.y * v.y + v.z * v.z + v.w * v.w;
    }
    lds_msq[c] = s;
  }

  // ---------------- stage tile 0 (clusters 0..31) into buffer 0 -----------
  const int sr = tid >> 3;          // staging row 0..31
  const int sd = (tid & 7) * 8;     // dim offset 0..56
  {
    const float* mp = mhead + (size_t)sr * DIMH + sd;
    float4 v0 = *(const float4*)mp;
    float4 v1 = *(const float4*)(mp + 4);
    float* l = &lds_b[0][sr * LSTRIDE + sd];
    l[0] = v0.x; l[1] = v0.y; l[2] = v0.z; l[3] = v0.w;
    l[4] = v1.x; l[5] = v1.y; l[6] = v1.z; l[7] = v1.w;
  }
  __syncthreads();

  float bestv[8];
  int   bidx[8];
#pragma unroll
  for (int j = 0; j < 8; ++j) { bestv[j] = 3.4e38f; bidx[j] = 0; }

  // ---------------- main loop: 32 iterations x 32 clusters -----------------
  for (int t = 0; t < NT; ++t) {
    const int buf = t & 1;
    if (t + 1 < NT) {               // stage next tile into the other buffer
      const float* mp = mhead + (size_t)((t + 1) * CT + sr) * DIMH + sd;
      float4 v0 = *(const float4*)mp;
      float4 v1 = *(const float4*)(mp + 4);
      float* l = &lds_b[buf ^ 1][sr * LSTRIDE + sd];
      l[0] = v0.x; l[1] = v0.y; l[2] = v0.z; l[3] = v0.w;
      l[4] = v1.x; l[5] = v1.y; l[6] = v1.z; l[7] = v1.w;
    }
    if (t + 2 < NT)                 // warm L2/WGP$ for the tile after next
      __builtin_prefetch(mhead + (size_t)((t + 2) * CT + sr) * DIMH + sd, 0, 3);

    // Two independent 16-column subtiles -> two independent WMMA chains.
    v8f acc[2];
#pragma unroll
    for (int u = 0; u < 2; ++u) {
      // B 4x16 layout: lane l -> column (cluster) l&15; K rows {0,1}/{2,3}
      // per half-wave -> contiguous float2 per K-chunk.
      const float* lb = &lds_b[buf][(u * 16 + nlo) * LSTRIDE + hi * 2];
      v2f bf[16];
#pragma unroll
      for (int kk = 0; kk < 16; ++kk) {       // preload: one DS wait total
        v2f b; b.x = lb[kk * 4]; b.y = lb[kk * 4 + 1];
        bf[kk] = b;
      }
      v8f a = {};
#pragma unroll
      for (int kk = 0; kk < 16; ++kk)         // back-to-back WMMA chain
        a = __builtin_amdgcn_wmma_f32_16x16x4_f32(
            /*neg_a=*/false, a_frag[kk], /*neg_b=*/false, bf[kk],
            /*c_mod=*/(short)0, a, /*reuse_a=*/false, /*reuse_b=*/false);
      acc[u] = a;
    }

    // argmin update: dist = ||m||^2 - 2*x.m  (||x||^2 dropped, const in k)
#pragma unroll
    for (int u = 0; u < 2; ++u) {
      const int c = t * CT + u * 16 + nlo;
      const float msq = lds_msq[c];
#pragma unroll
      for (int j = 0; j < 8; ++j) {
        float dist = msq - 2.0f * acc[u][j];
        if (dist < bestv[j]) { bestv[j] = dist; bidx[j] = c; }
      }
    }
    __syncthreads();                // protects double-buffered tiles
  }

  // ---------------- argmin butterfly over the 16 N-lanes -------------------
#pragma unroll
  for (int j = 0; j < 8; ++j) {
    float v = bestv[j]; int i = bidx[j];
#pragma unroll
    for (int m = 8; m >= 1; m >>= 1) {
      float ov = __shfl_xor(v, m, 32);
      int   oi = __shfl_xor(i, m, 32);
      if (ov < v || (ov == v && oi < i)) { v = ov; i = oi; }  // first-occurrence tie-break
    }
    bestv[j] = v; bidx[j] = i;
  }

  if (nlo == 0) {
#pragma unroll
    for (int j = 0; j < 8; ++j) {
      const int tok = token0 + j + hi * 8;    // flat (b*n) token index
      out[(size_t)tok * NHEADS + head] = (float)bidx[j];
    }
  }
}

extern "C" void kernel_launch(void* const* d_in, const int* in_sizes, int n_in,
                              void* d_out, int out_size, void* d_ws, size_t ws_size,
                              hipStream_t stream) {
  const float* x     = (const float*)d_in[0];
  const float* means = (const float*)d_in[1];
  float* out = (float*)d_out;

  const int tokens = in_sizes[0] / (NHEADS * DIMH);   // b*n = 8192
  dim3 grid(tokens / (TOKTILE * WAVES), NHEADS);      // (64, 8)
  vq_argmin_kernel<<<grid, 256, 0, stream>>>(x, means, out);
}